// CenterCTCLoss_87600152969910
// MI455X (gfx1250) — compile-verified
//
#include <hip/hip_runtime.h>
#include <stdint.h>

#define NC      85
#define BLANK   84
#define TT      96
#define DD      64
#define ALPHA_C 0.05f
#define ROWS    8          // batch rows per workgroup

// ---------------------------------------------------------------------------
// CDNA5 async global->LDS copy (ASYNCcnt path, gfx1250)
// ---------------------------------------------------------------------------
__device__ __forceinline__ void asyncLoadB128ToLds(void* ldsDst, const void* gSrc) {
    uint32_t loff  = (uint32_t)(uintptr_t)ldsDst;   // low 32 bits of LDS aperture = LDS offset
    uint64_t gaddr = (uint64_t)(uintptr_t)gSrc;
    asm volatile("global_load_async_to_lds_b128 %0, %1, off"
                 :: "v"(loff), "v"(gaddr) : "memory");
}
__device__ __forceinline__ void waitAsync() {
    asm volatile("s_wait_asynccnt 0x0" ::: "memory");
}

// ---------------------------------------------------------------------------
// Kernel 0: zero workspace accumulators (ws is poisoned by harness)
// ---------------------------------------------------------------------------
__global__ __launch_bounds__(256) void k_init(unsigned int* __restrict__ counts,
                                              float* __restrict__ lossAcc,
                                              float* __restrict__ fsum) {
    int i = blockIdx.x * blockDim.x + threadIdx.x;
    if (i < NC) counts[i] = 0u;
    if (i == 0) *lossAcc = 0.0f;
    if (i < NC * DD) fsum[i] = 0.0f;
}

// ---------------------------------------------------------------------------
// Kernel 1: per-row argmax + CTC mask + class counts + loss + masked feature
// sums. One block (256 thr = 8 waves) handles ROWS batch rows; LDS accumulators
// flushed to global once per block.
// ---------------------------------------------------------------------------
__global__ __launch_bounds__(256) void k_main(const float* __restrict__ preds,
                                              const float* __restrict__ feats,
                                              const int*   __restrict__ labels,
                                              const float* __restrict__ centers,
                                              int B,
                                              unsigned int* __restrict__ gCounts,
                                              float* __restrict__ gLoss,
                                              float* __restrict__ gS) {
    __shared__ __attribute__((aligned(16))) float sC[NC * DD];  // centers copy
    __shared__ float         sS[NC * DD];                       // masked feature sums
    __shared__ unsigned int  sCnt[NC];
    __shared__ int           sPred[TT];
    __shared__ int           sLab[TT];
    __shared__ unsigned char sMask[TT];
    __shared__ float         sRed[8];

    const int tid  = threadIdx.x;
    const int lane = tid & 31;
    const int wave = tid >> 5;

    // async broadcast of centers table into LDS (overlaps first argmax phase)
    for (int i = tid; i < (NC * DD) / 4; i += 256)
        asyncLoadB128ToLds(&sC[i * 4], centers + i * 4);

    for (int i = tid; i < NC * DD; i += 256) sS[i] = 0.0f;
    if (tid < NC) sCnt[tid] = 0u;

    float lossAcc = 0.0f;

    for (int r = 0; r < ROWS; ++r) {
        const int b = blockIdx.x * ROWS + r;
        if (b >= B) break;                          // uniform across block

        __syncthreads();                            // sPred/sLab/sMask reuse guard
        if (tid < TT) sLab[tid] = labels[(size_t)b * TT + tid];

        // argmax over 85 classes: one wave per timestep, 12 timesteps per wave
        const float* predsB = preds + (size_t)b * TT * NC;
        for (int t = wave * 12; t < wave * 12 + 12; ++t) {
            const float* row = predsB + t * NC;
            float best = row[lane];
            int   bi   = lane;
            float v1 = row[lane + 32];
            if (v1 > best) { best = v1; bi = lane + 32; }
            if (lane < NC - 64) {
                float v2 = row[lane + 64];
                if (v2 > best) { best = v2; bi = lane + 64; }
            }
            // wave32 butterfly reduce, first-occurrence tie-break (min index)
            for (int off = 16; off > 0; off >>= 1) {
                float ov = __shfl_xor(best, off, 32);
                int   oi = __shfl_xor(bi,   off, 32);
                if (ov > best || (ov == best && oi < bi)) { best = ov; bi = oi; }
            }
            if (lane == 0) sPred[t] = bi;
        }

        if (r == 0) waitAsync();                    // centers landed in LDS
        __syncthreads();

        // CTC collapse mask + per-class counts
        if (tid < TT) {
            int  pl = sPred[tid];
            bool m  = (pl != BLANK) && (tid == TT - 1 || pl != sPred[tid + 1]);
            sMask[tid] = m ? 1 : 0;
            if (m) atomicAdd(&sCnt[sLab[tid]], 1u);
        }
        __syncthreads();

        // masked loss + masked feature sums (float4 streaming loads)
        const float4* f4 = (const float4*)(feats + (size_t)b * TT * DD);
        for (int i = tid; i < (TT * DD) / 4; i += 256) {   // 6 iterations
            int t = i >> 4;
            if (sMask[t]) {
                int    lab = sLab[t];
                int    d4  = (i & 15) * 4;
                float4 f   = f4[i];
                const float* c = &sC[lab * DD + d4];
                float dx = c[0] - f.x, dy = c[1] - f.y;
                float dz = c[2] - f.z, dw = c[3] - f.w;
                lossAcc += dx * dx + dy * dy + dz * dz + dw * dw;
                float* s = &sS[lab * DD + d4];
                atomicAdd(&s[0], f.x);              // ds_add_f32
                atomicAdd(&s[1], f.y);
                atomicAdd(&s[2], f.z);
                atomicAdd(&s[3], f.w);
            }
        }
    }

    __syncthreads();

    // flush block accumulators to global (once per ROWS rows)
    if (tid < NC && sCnt[tid] != 0u) atomicAdd(&gCounts[tid], sCnt[tid]);
    for (int i = tid; i < NC * DD; i += 256) {
        float v = sS[i];
        if (v != 0.0f) atomicAdd(&gS[i], v);
    }
    for (int off = 16; off > 0; off >>= 1) lossAcc += __shfl_xor(lossAcc, off, 32);
    if (lane == 0) sRed[wave] = lossAcc;
    __syncthreads();
    if (tid == 0) {
        float s = 0.0f;
        for (int w = 0; w < 8; ++w) s += sRed[w];
        atomicAdd(gLoss, 0.5f * s);
    }
}

// ---------------------------------------------------------------------------
// Kernel 2: finalize: upd[c,d] = ALPHA/(1+cnt_c) * (cnt_c*centers - S)
// out = [loss, new_centers(85*64)]
// ---------------------------------------------------------------------------
__global__ __launch_bounds__(256) void k_final(const float* __restrict__ centers,
                                               const float* __restrict__ gS,
                                               const unsigned int* __restrict__ gCounts,
                                               const float* __restrict__ gLoss,
                                               float* __restrict__ out) {
    int i = blockIdx.x * blockDim.x + threadIdx.x;
    if (i == 0) out[0] = *gLoss;
    if (i < NC * DD) {
        int   c     = i >> 6;                       // DD == 64
        float cnt   = (float)gCounts[c];
        float scale = ALPHA_C / (1.0f + cnt);
        float cc    = centers[i];
        float upd   = scale * (cnt * cc - gS[i]);
        out[1 + i]  = cc - upd;
    }
}

// ---------------------------------------------------------------------------
extern "C" void kernel_launch(void* const* d_in, const int* in_sizes, int n_in,
                              void* d_out, int out_size, void* d_ws, size_t ws_size,
                              hipStream_t stream) {
    const float* preds   = (const float*)d_in[0];   // [B,T,85]
    const float* feats   = (const float*)d_in[1];   // [B,T,64]
    const int*   labels  = (const int*)d_in[2];     // [B,T]
    const float* centers = (const float*)d_in[3];   // [85,64]
    float* out = (float*)d_out;

    const int B = in_sizes[2] / TT;                 // 4096

    // workspace layout (~23 KB)
    uint8_t* ws = (uint8_t*)d_ws;
    unsigned int* counts  = (unsigned int*)ws;          // 85 u32
    float*        lossAcc = (float*)(ws + 512);         // 1 f32
    float*        fsum    = (float*)(ws + 1024);        // 5440 f32

    k_init <<<(NC * DD + 255) / 256, 256, 0, stream>>>(counts, lossAcc, fsum);
    k_main <<<(B + ROWS - 1) / ROWS, 256, 0, stream>>>(preds, feats, labels, centers,
                                                       B, counts, lossAcc, fsum);
    k_final<<<(NC * DD + 1 + 255) / 256, 256, 0, stream>>>(centers, fsum, counts,
                                                           lossAcc, out);
}